// RadiusConnect_33036888441075
// MI455X (gfx1250) — compile-verified
//
#include <hip/hip_runtime.h>

// CDNA5 radius-graph using V_WMMA_F32_16X16X4_F32.
// dist2(M,N) = |d_M|^2 + |s_N|^2 - 2 d_M.s_N computed as D = A x B + C with
//   A(M,:) = (-2x_d, -2y_d, -2z_d, 1)    [16x4 f32, 2 VGPRs]
//   B(:,N) = ( x_s,   y_s,   z_s, |s|^2) [4x16 f32, 2 VGPRs]
//   C(M,N) = |d_M|^2                     [16x16 f32, 8 VGPRs]

typedef __attribute__((ext_vector_type(2))) float v2f;
typedef __attribute__((ext_vector_type(8))) float v8f;

#define RADIUS2 0.04f   // R = 0.2
#define MAXNB   32

// gfx1250 is wave32. ballot(bare_compare) folds into a single v_cmp with SGPR
// dest; composite predicates do NOT fold (backend re-materializes), so we
// ballot each compare separately and AND the masks scalar-side.
#define WAVE_BALLOT(p) __builtin_amdgcn_ballot_w32(p)

// lower_bound over low 32-bit words of a sorted int64 array (values 0..7)
__device__ __forceinline__ int lbound_lo32(const int* __restrict__ b2, int n, int key) {
  int lo = 0, hi = n;
  while (lo < hi) {
    int mid = (lo + hi) >> 1;
    if (b2[2 * mid] < key) lo = mid + 1; else hi = mid;
  }
  return lo;
}

__global__ void __launch_bounds__(256)
fill_neg1_kernel(int* __restrict__ out, int n) {
  int i = blockIdx.x * blockDim.x + threadIdx.x;
  if (i < n) out[i] = -1;
}

__global__ void __launch_bounds__(256)
radius_graph_wmma_kernel(const float* __restrict__ xs,  // [n_src,3]
                         const int*   __restrict__ bs2, // int64 batch_src as int pairs
                         const float* __restrict__ xd,  // [n_dst,3]
                         const int*   __restrict__ bd2, // int64 batch_dst as int pairs
                         int* __restrict__ out,         // [2 * n_dst * MAXNB]
                         int n_src, int n_dst) {
  const int lane    = threadIdx.x & 31;
  const int wave    = threadIdx.x >> 5;
  const int dstBase = (blockIdx.x * (blockDim.x >> 5) + wave) << 4;
  if (dstBase >= n_dst) return;            // wave-uniform

  const int  row = lane & 15;
  const bool hiH = lane >= 16;
  const int  hi8 = hiH ? 8 : 0;
  const int  myDst = dstBase + row;

  // ---- per-row dst data (lane&15 <-> row; hi/lo halves duplicate) ----
  const float dx = xd[myDst * 3 + 0];
  const float dy = xd[myDst * 3 + 1];
  const float dz = xd[myDst * 3 + 2];
  const float d2 = dx * dx + dy * dy + dz * dz;
  const int   myB = bd2[2 * myDst];

  // A matrix: lanes 0-15 hold K=0,1 ; lanes 16-31 hold K=2,3
  v2f a;
  a.x = hiH ? (-2.0f * dz) : (-2.0f * dx);
  a.y = hiH ? 1.0f         : (-2.0f * dy);

  // C matrix (|d|^2 per row) + per-row batch, per the 16x16 f32 C/D layout:
  // VGPR r holds row (r + 8*(lane>=16))
  v8f c;
  int rowB[8];
#pragma unroll
  for (int r = 0; r < 8; ++r) {
    c[r]    = __shfl(d2, r + hi8, 32);
    rowB[r] = __shfl(myB, r + hi8, 32);
  }

  // loop-invariant per-lane row-ownership flags (lane l<16 owns row l)
  bool isRow[8];
#pragma unroll
  for (int r = 0; r < 8; ++r) isRow[r] = (row & 7) == r;
  const int mShift = (row & 8) * 2;   // 0 for rows 0-7, 16 for rows 8-15

  // ---- batch segment of src for each row; wave union ----
  int st = lbound_lo32(bs2, n_src, myB);
  int en = lbound_lo32(bs2, n_src, myB + 1);
#pragma unroll
  for (int off = 16; off; off >>= 1) {
    st = min(st, __shfl_xor(st, off, 32));
    en = max(en, __shfl_xor(en, off, 32));
  }

  int count = 0;                 // neighbors emitted for row==lane (lanes 0-15)
  const int col = lane & 15;     // this lane's column in the 16x16 tile

  for (int tb = (st & ~15); tb < en; tb += 16) {
    // early exit once every row has MAXNB neighbors. Lanes 0-15 own rows and
    // are exactly the low 16 ballot bits; hi lanes never emit (count==0), so
    // mask them with the constant instead of a second compare.
    if ((WAVE_BALLOT(count < MAXNB) & 0xFFFFu) == 0u) break;

    const int   s  = tb + col;               // n_src % 16 == 0 -> in-bounds
    const float sx = xs[s * 3 + 0];
    const float sy = xs[s * 3 + 1];
    const float sz = xs[s * 3 + 2];
    const float s2 = sx * sx + sy * sy + sz * sz;
    const int   sb = bs2[2 * s];

    // B matrix: lanes 0-15 hold K=0,1 ; lanes 16-31 hold K=2,3
    v2f b;
    b.x = hiH ? sz : sx;
    b.y = hiH ? s2 : sy;

    // D = A*B + C  : full 16x16 tile of dist2
    v8f dmat = __builtin_amdgcn_wmma_f32_16x16x4_f32(
        false, a, false, b, (short)0, c, false, false);

    // ballot bit layout == C/D layout: bits[15:0]=row r, bits[31:16]=row r+8.
    // Ballot each bare compare (folds to v_cmp->SGPR), AND scalar-side.
    unsigned bl[8];
#pragma unroll
    for (int r = 0; r < 8; ++r) {
      unsigned rad = WAVE_BALLOT(dmat[r] <= RADIUS2);
      unsigned beq = WAVE_BALLOT(sb == rowB[r]);
      bl[r] = rad & beq;
    }

    // lane l (0..15) owns row l: flat cndmask chain (no divergent branches)
    unsigned bm = 0;
#pragma unroll
    for (int r = 0; r < 8; ++r) bm = isRow[r] ? bl[r] : bm;

    unsigned m = (bm >> mShift) & 0xFFFFu;
    if (hiH) m = 0;

    // emit ascending src indices -> matches reference's lowest-index-first rule
    while (m && count < MAXNB) {
      int bit = __builtin_ctz(m);
      m &= m - 1;
      int si = tb + bit;
      out[myDst * MAXNB + count] = si;                       // edge_src
      out[n_dst * MAXNB + myDst * MAXNB + count] = myDst;    // edge_dst
      ++count;
    }
  }
}

extern "C" void kernel_launch(void* const* d_in, const int* in_sizes, int n_in,
                              void* d_out, int out_size, void* d_ws, size_t ws_size,
                              hipStream_t stream) {
  const float* xs  = (const float*)d_in[0];  // node_coord_src [N,3] f32
  const int*   bs2 = (const int*)  d_in[1];  // batch_src      [N]   i64
  const float* xd  = (const float*)d_in[2];  // node_coord_dst [N,3] f32
  const int*   bd2 = (const int*)  d_in[3];  // batch_dst      [N]   i64
  int* out = (int*)d_out;                    // edge_src ++ edge_dst (int32)

  const int n_src = in_sizes[0] / 3;
  const int n_dst = in_sizes[2] / 3;

  // pad everything with -1 first; main kernel only writes found edges
  fill_neg1_kernel<<<(out_size + 255) / 256, 256, 0, stream>>>(out, out_size);

  const int wavesNeeded   = (n_dst + 15) / 16;   // one wave per 16 dst rows
  const int wavesPerBlock = 8;                   // 256 threads = 8 wave32
  const int blocks = (wavesNeeded + wavesPerBlock - 1) / wavesPerBlock;
  radius_graph_wmma_kernel<<<blocks, 256, 0, stream>>>(xs, bs2, xd, bd2, out,
                                                       n_src, n_dst);
}